// VectorQuantizer2_41248865910958
// MI455X (gfx1250) — compile-verified
//
#include <hip/hip_runtime.h>
#include <hip/hip_bf16.h>
#include <math.h>

typedef float v2f __attribute__((ext_vector_type(2)));
typedef float v4f __attribute__((ext_vector_type(4)));
typedef float v8f __attribute__((ext_vector_type(8)));
typedef unsigned int v4u __attribute__((ext_vector_type(4)));
typedef int v4i __attribute__((ext_vector_type(4)));
typedef int v8i __attribute__((ext_vector_type(8)));

#define NROWS 16384          // 16*32*32 flattened spatial positions
#define KDIM  256            // embedding dim
#define NE    8192           // codebook entries
#define HWD   1024           // 32*32
#define CHW   (KDIM * HWD)   // 262144
#define NUMEL (NROWS * KDIM) // 4194304
#define BETA_F 0.25f

#define TILE_CODES 32        // codes per tile iteration (2 WMMA N-tiles)
#define NT (NE / TILE_CODES) // 256 tile iterations
#define LDS_STRIDE 260       // 256 + 4 pad floats: conflict-free b64 reads
#define BUF_FLOATS (TILE_CODES * LDS_STRIDE)

#if __has_builtin(__builtin_amdgcn_tensor_load_to_lds)
#define USE_TDM 1
#else
#define USE_TDM 0
#endif

// ---------------------------------------------------------------------------
// Kernel 1: codebook squared norms; also zero the loss accumulator slot.
// ---------------------------------------------------------------------------
__global__ __launch_bounds__(256) void vq_norms(const float* __restrict__ cb,
                                                float* __restrict__ norms,
                                                float* __restrict__ lossSlot) {
    int j = blockIdx.x * blockDim.x + threadIdx.x;
    if (j == 0) *lossSlot = 0.0f;
    if (j < NE) {
        const v4f* row = (const v4f*)(cb + (size_t)j * KDIM);
        float s = 0.0f;
#pragma unroll 8
        for (int i = 0; i < KDIM / 4; ++i) {
            v4f v = row[i];
            s += v.x * v.x + v.y * v.y + v.z * v.z + v.w * v.w;
        }
        norms[j] = s;
    }
}

#if USE_TDM
// Issue a TDM load of TILE_CODES x 256 f32 codebook rows into LDS.
// D# pad feature inserts 4 dwords after every 256 dwords -> LDS row stride 260.
__device__ __forceinline__ void tdm_load_tile(const float* gsrc, unsigned lds_off) {
    const unsigned long long ga = (unsigned long long)(const void*)gsrc;
    v4u g0;
    g0[0] = 1u;                                          // count=1, user desc
    g0[1] = lds_off;                                     // lds_addr (bytes)
    g0[2] = (unsigned)(ga & 0xffffffffu);                // global_addr[31:0]
    g0[3] = (unsigned)((ga >> 32) & 0x01ffffffu)         // global_addr[56:32]
            | 0x80000000u;                               // type=2 ("image")
    v8i g1;
    g1[0] = (int)(0x20000u                               // data_size=2 (4B)
                  | 0x100000u                            // pad_enable
                  | (7u << 22)                           // pad_interval: 256 dw
                  | (3u << 25));                         // pad_amount: 4 dw
    g1[1] = (int)((unsigned)KDIM << 16);                 // tensor_dim0=256 (lo16)
    g1[2] = (int)((unsigned)TILE_CODES << 16);           // tensor_dim1=32 (lo16)
    g1[3] = (int)((unsigned)KDIM << 16);                 // tile_dim0=256
    g1[4] = TILE_CODES;                                  // tile_dim1=32, tile_dim2=0
    g1[5] = KDIM;                                        // tensor_dim0_stride=256
    g1[6] = 0;
    g1[7] = 0;
    const v4i z4 = {0, 0, 0, 0};                         // groups 2/3: unused (2D)
    const v8i z8 = {0, 0, 0, 0, 0, 0, 0, 0};             // extra group: unused
    __builtin_amdgcn_tensor_load_to_lds(g0, g1, z4, z4, z8, 0);
}
#endif

// ---------------------------------------------------------------------------
// Kernel 2: fused distance-GEMM + argmin.
// Block = 256 threads = 8 waves; wave w owns rows [blockBase + 16w, +16).
// A panel (16 rows x 256 K, fp32) resident in VGPRs in WMMA 16x16x4 A-layout.
// Codebook streamed in 32-code tiles: TDM -> double-buffered LDS; two
// independent WMMA accumulator chains per wave.
// ---------------------------------------------------------------------------
__global__ __launch_bounds__(256) void vq_argmin(const float* __restrict__ z,
                                                 const float* __restrict__ cb,
                                                 const float* __restrict__ norms,
                                                 int* __restrict__ idxOut,
                                                 float* __restrict__ idxF) {
    __shared__ float ldsB[2][BUF_FLOATS];

    const int tid  = threadIdx.x;
    const int lane = tid & 31;          // wave32
    const int wave = tid >> 5;
    const int rowBase = blockIdx.x * 128 + wave * 16;

    // ---- Resident A fragments: lane l holds row (l&15), K = 4kb + 2*(l>>4) + {0,1}
    const int mrow = rowBase + (lane & 15);
    const int bI   = mrow >> 10;        // tiles never cross batch (1024 % 16 == 0)
    const int hw   = mrow & 1023;
    const float* zbase = z + (size_t)bI * CHW + hw;   // c-stride = HWD
    const int kh = (lane >> 4) << 1;    // 0 for lanes 0-15, 2 for lanes 16-31

    v2f areg[64];
#pragma unroll
    for (int kb = 0; kb < 64; ++kb) {
        const int k0 = kb * 4 + kh;
        v2f a;
        a.x = zbase[(size_t)k0 * HWD];
        a.y = zbase[(size_t)(k0 + 1) * HWD];
        areg[kb] = a;
    }

    float best[8];
    int   bidx[8];
#pragma unroll
    for (int r = 0; r < 8; ++r) { best[r] = __builtin_inff(); bidx[r] = 0; }

    const int col = lane & 15;          // WMMA N index within a 16-code tile

#if USE_TDM
    const unsigned ldsOff0 = (unsigned)(unsigned long long)(const void*)&ldsB[0][0];
    const unsigned ldsOff1 = (unsigned)(unsigned long long)(const void*)&ldsB[1][0];
    if (wave == 0) tdm_load_tile(cb, ldsOff0);          // prologue: tile 0
#endif

    for (int ct = 0; ct < NT; ++ct) {
        const int codeBase = ct * TILE_CODES;

#if USE_TDM
        if (wave == 0) __builtin_amdgcn_s_wait_tensorcnt(0);
        __syncthreads();   // tile ct visible to all; all done reading buf[(ct+1)&1]
        if (wave == 0 && ct + 1 < NT)
            tdm_load_tile(cb + (size_t)(codeBase + TILE_CODES) * KDIM,
                          ((ct + 1) & 1) ? ldsOff1 : ldsOff0);
        const float* bbase = &ldsB[ct & 1][0];
#else
        __syncthreads();
        {   // cooperative stage: 32 contiguous codebook rows (32KB) -> LDS (padded)
            const v4f* src = (const v4f*)(cb + (size_t)codeBase * KDIM);
#pragma unroll
            for (int i = 0; i < 8; ++i) {
                const int f4   = tid + 256 * i;      // 0..2047 float4s
                const int rrow = f4 >> 6;            // 64 float4 per codebook row
                const int kf   = (f4 & 63) << 2;
                *(v4f*)&ldsB[0][rrow * LDS_STRIDE + kf] = src[f4];
            }
        }
        __syncthreads();
        const float* bbase = &ldsB[0][0];
#endif

        // issue norm loads early so they overlap the WMMA chain
        const float sn0 = norms[codeBase + col];
        const float sn1 = norms[codeBase + 16 + col];

        // ---- two independent 16x16 accumulation chains over K = 256
        v8f acc0 = (v8f){0.f, 0.f, 0.f, 0.f, 0.f, 0.f, 0.f, 0.f};
        v8f acc1 = (v8f){0.f, 0.f, 0.f, 0.f, 0.f, 0.f, 0.f, 0.f};
        const float* brow0 = bbase + col * LDS_STRIDE + kh;
        const float* brow1 = bbase + (col + 16) * LDS_STRIDE + kh;
#pragma unroll
        for (int kb = 0; kb < 64; ++kb) {
            v2f b0 = *(const v2f*)(brow0 + kb * 4);
            v2f b1 = *(const v2f*)(brow1 + kb * 4);
            acc0 = __builtin_amdgcn_wmma_f32_16x16x4_f32(
                false, areg[kb], false, b0, (short)0, acc0, false, false);
            acc1 = __builtin_amdgcn_wmma_f32_16x16x4_f32(
                false, areg[kb], false, b1, (short)0, acc1, false, false);
        }

        // ---- score = ||e||^2 - 2 x.e ; strict < keeps lowest index
        const int j0 = codeBase + col;
        const int j1 = codeBase + 16 + col;
#pragma unroll
        for (int r = 0; r < 8; ++r) {
            const float s0 = fmaf(-2.0f, acc0[r], sn0);
            if (s0 < best[r]) { best[r] = s0; bidx[r] = j0; }
            const float s1 = fmaf(-2.0f, acc1[r], sn1);
            if (s1 < best[r]) { best[r] = s1; bidx[r] = j1; }
        }
    }

    // ---- reduce (min, idx) across the 16 lanes that share a row
#pragma unroll
    for (int r = 0; r < 8; ++r) {
        float v  = best[r];
        int   bi = bidx[r];
#pragma unroll
        for (int m = 8; m >= 1; m >>= 1) {
            const float ov = __shfl_xor(v, m, 16);
            const int   oi = __shfl_xor(bi, m, 16);
            if (ov < v || (ov == v && oi < bi)) { v = ov; bi = oi; }
        }
        if ((lane & 15) == 0) {
            // lanes 0-15 hold M = r, lanes 16-31 hold M = r + 8
            const int row = rowBase + r + ((lane >> 4) << 3);
            idxOut[row] = bi;
            idxF[row]   = (float)bi;
        }
    }
}

// ---------------------------------------------------------------------------
// Kernel 3: gather quantized vectors (output shares z's [B,C,H,W] linear
// layout) and accumulate loss = (1+beta) * mean((q - z)^2).
// ---------------------------------------------------------------------------
__global__ __launch_bounds__(256) void vq_gather_loss(const float* __restrict__ z,
                                                      const float* __restrict__ cb,
                                                      const int* __restrict__ idx,
                                                      float* __restrict__ out,
                                                      float* __restrict__ lossSlot) {
    const size_t i = (size_t)blockIdx.x * 256 + threadIdx.x;
    const int b  = (int)(i >> 18);          // / CHW
    const int c  = (int)((i >> 10) & 255);  // channel
    const int hw = (int)(i & 1023);
    const int n  = (b << 10) + hw;          // flattened row id
    const float q  = cb[(size_t)idx[n] * KDIM + c];
    const float zv = z[i];
    out[i] = q;
    const float d = q - zv;
    float p = d * d;
#pragma unroll
    for (int m = 16; m >= 1; m >>= 1) p += __shfl_xor(p, m, 32);
    if ((threadIdx.x & 31) == 0)
        atomicAdd(lossSlot, p * ((1.0f + BETA_F) / (float)NUMEL));
}

// ---------------------------------------------------------------------------
extern "C" void kernel_launch(void* const* d_in, const int* in_sizes, int n_in,
                              void* d_out, int out_size, void* d_ws, size_t ws_size,
                              hipStream_t stream) {
    const float* z  = (const float*)d_in[0];   // [16,256,32,32]
    const float* cb = (const float*)d_in[1];   // [8192,256]
    float* out      = (float*)d_out;           // quant | loss | indices(float)
    float* lossSlot = out + NUMEL;
    float* idxF     = out + NUMEL + 1;

    float* norms = (float*)d_ws;                                   // 8192 f32
    int*   idxI  = (int*)((char*)d_ws + NE * sizeof(float));       // 16384 i32

    vq_norms      <<<NE / 256,    256, 0, stream>>>(cb, norms, lossSlot);
    vq_argmin     <<<NROWS / 128, 256, 0, stream>>>(z, cb, norms, idxI, idxF);
    vq_gather_loss<<<NUMEL / 256, 256, 0, stream>>>(z, cb, idxI, out, lossSlot);
}